// PPCA_with_ARD_model_55121610276952
// MI455X (gfx1250) — compile-verified
//
#include <hip/hip_runtime.h>
#include <hip/hip_bf16.h>
#include <math.h>

#define D_DIM 4096
#define K_DIM 128
#define KP    144      // 129 (K+1) padded to 9*16
#define N_DIM 8192
#define EPS_C 1e-6f

typedef float v2f __attribute__((ext_vector_type(2)));
typedef float v8f __attribute__((ext_vector_type(8)));

__device__ __forceinline__ v8f wmma_f32(v2f a, v2f b, v8f c) {
    // D = A(16x4,f32) * B(4x16,f32) + C(16x16,f32)
    return __builtin_amdgcn_wmma_f32_16x16x4_f32(
        /*neg_a=*/false, a, /*neg_b=*/false, b,
        /*c_mod=*/(short)0, c, /*reuse_a=*/false, /*reuse_b=*/false);
}

// Async DMA: global -> LDS, 16B per lane, tracked by ASYNCcnt.
__device__ __forceinline__ void async_load_b128(unsigned lds_addr, const void* gptr) {
    asm volatile("global_load_async_to_lds_b128 %0, %1, off"
                 :: "v"(lds_addr), "v"(gptr) : "memory");
}
__device__ __forceinline__ void wait_async0() {
    asm volatile("s_wait_asynccnt 0" ::: "memory");
}

// ---------------------------------------------------------------------------
// K1: V[d][j] = U[d][j] / sigma[d],  U = [W*alpha, sqrt(eps)*1, 0-pad]
// ---------------------------------------------------------------------------
__global__ __launch_bounds__(256) void k_build_V(const float* __restrict__ theta,
                                                 float* __restrict__ V,
                                                 float* __restrict__ invsig) {
    int idx = blockIdx.x * 256 + threadIdx.x;
    if (idx >= D_DIM * KP) return;
    int d = idx / KP, j = idx - d * KP;
    float sig = theta[D_DIM * K_DIM + D_DIM + d];
    float inv = 1.0f / sig;
    float u;
    if (j < K_DIM)       u = theta[d * K_DIM + j] * theta[D_DIM * K_DIM + 2 * D_DIM + j];
    else if (j == K_DIM) u = 1.0e-3f;   // sqrt(1e-6)
    else                 u = 0.0f;
    V[idx] = u * inv;
    if (j == 0) invsig[d] = inv;
}

// ---------------------------------------------------------------------------
// K1b: tmu[i] = sum_d mu[d] * V[d][i]   (for the T = X@V - tmu decomposition)
// ---------------------------------------------------------------------------
__global__ __launch_bounds__(256) void k_tmu(const float* __restrict__ theta,
                                             const float* __restrict__ V,
                                             float* __restrict__ tmu) {
    int i = threadIdx.x;
    if (i >= KP) return;
    const float* mu = theta + D_DIM * K_DIM;
    float s = 0.f;
    for (int d = 0; d < D_DIM; ++d) s += mu[d] * V[(size_t)d * KP + i];
    tmu[i] = s;
}

// ---------------------------------------------------------------------------
// K2: M = I + sum_d sigma[d] * V[d,:]^T V[d,:]   (144x144, 81 WMMA tiles)
// ---------------------------------------------------------------------------
__global__ __launch_bounds__(128) void k_build_M(const float* __restrict__ theta,
                                                 const float* __restrict__ V,
                                                 float* __restrict__ M) {
    int wave = threadIdx.x >> 5;
    int lane = threadIdx.x & 31;
    int tile = blockIdx.x * 4 + wave;
    if (tile >= 81) return;                       // wave-uniform: EXEC stays full
    int ti = (tile / 9) * 16, tj = (tile % 9) * 16;
    const float* sigma = theta + D_DIM * K_DIM + D_DIM;
    int half = lane >> 4, lan = lane & 15;

    v8f acc = {0.f, 0.f, 0.f, 0.f, 0.f, 0.f, 0.f, 0.f};
    for (int d = 0; d < D_DIM; d += 4) {
        int k = d + 2 * half;
        float s0 = sigma[k], s1 = sigma[k + 1];
        v2f a, b;
        a.x = s0 * V[(size_t)k * KP + ti + lan];          // A: M=lan, K=k
        a.y = s1 * V[(size_t)(k + 1) * KP + ti + lan];    //    K=k+1
        b.x = V[(size_t)k * KP + tj + lan];               // B: N=lan, K=k
        b.y = V[(size_t)(k + 1) * KP + tj + lan];
        acc = wmma_f32(a, b, acc);
    }
    #pragma unroll
    for (int r = 0; r < 8; ++r) {
        int i = ti + r + 8 * half;                        // C/D: VGPR r -> M=r (+8 hi half)
        int j = tj + lan;
        M[i * KP + j] = acc[r] + ((i == j) ? 1.0f : 0.0f);
    }
}

// ---------------------------------------------------------------------------
// K3: single-block Cholesky of M (in-place, global), logdet, explicit M^{-1}
// ---------------------------------------------------------------------------
__global__ __launch_bounds__(256) void k_chol_inv(const float* __restrict__ theta,
                                                  float* __restrict__ M,
                                                  float* __restrict__ Minv,
                                                  float* __restrict__ scal) {
    int tid = threadIdx.x;
    for (int k = 0; k < KP; ++k) {
        if (tid == 0) M[k * KP + k] = sqrtf(M[k * KP + k]);
        __syncthreads();
        float dk = M[k * KP + k];
        for (int i = k + 1 + tid; i < KP; i += 256) M[i * KP + k] /= dk;
        __syncthreads();
        for (int i = k + 1 + tid; i < KP; i += 256) {
            float lik = M[i * KP + k];
            for (int j = k + 1; j <= i; ++j) M[i * KP + j] -= lik * M[j * KP + k];
        }
        __syncthreads();
    }
    // logdet(Sigma) = sum log sigma + 2 * sum log diag(L)
    __shared__ float red[256];
    float part = 0.f;
    for (int k = tid; k < KP; k += 256) part += logf(M[k * KP + k]);
    part *= 2.0f;
    const float* sigma = theta + D_DIM * K_DIM + D_DIM;
    for (int d = tid; d < D_DIM; d += 256) part += logf(sigma[d]);
    red[tid] = part; __syncthreads();
    for (int s = 128; s > 0; s >>= 1) { if (tid < s) red[tid] += red[tid + s]; __syncthreads(); }
    if (tid == 0) scal[0] = red[0];
    __syncthreads();
    // Minv: column c = tid, forward then back substitution with L
    if (tid < KP) {
        float y[KP];
        for (int i = 0; i < KP; ++i) {
            float s = (i == tid) ? 1.0f : 0.0f;
            for (int j = 0; j < i; ++j) s -= M[i * KP + j] * y[j];
            y[i] = s / M[i * KP + i];
        }
        for (int i = KP - 1; i >= 0; --i) {
            float s = y[i];
            for (int j = i + 1; j < KP; ++j) s -= M[j * KP + i] * y[j];
            y[i] = s / M[i * KP + i];
        }
        for (int i = 0; i < KP; ++i) Minv[i * KP + tid] = y[i];
    }
}

// ---------------------------------------------------------------------------
// K4: main streaming kernel. T = X @ V - tmu  and  s1[n] = sum (x-mu)^2/sigma.
// 128 blocks x 128 threads (4 waves); 64 rows x all 144 cols per block.
// Tiles staged via GLOBAL_LOAD_ASYNC_TO_LDS_B128 (ASYNCcnt), WMMA f32 16x16x4.
// ---------------------------------------------------------------------------
__global__ __launch_bounds__(128) void k_gemm(const float* __restrict__ x,
                                              const float* __restrict__ theta,
                                              const float* __restrict__ V,
                                              const float* __restrict__ invsig,
                                              const float* __restrict__ tmu,
                                              float* __restrict__ T,
                                              float* __restrict__ s1) {
    __shared__ float Xs[64][36];          // raw x tile; pitch 144B: 16B-aligned, conflict-free
    __shared__ float Vs[32 * KP];         // V K-tile (contiguous)
    __shared__ float s1p[128];
    const float* mu = theta + D_DIM * K_DIM;

    int tid = threadIdx.x;
    int lane = tid & 31, wave = tid >> 5;
    int half = lane >> 4, lan = lane & 15;
    int row0 = blockIdx.x * 64;
    int srow = tid >> 1;                  // 0..63: row for the s1 pass
    int cseg = (tid & 1) * 16;

    float s1acc = 0.f;
    v8f acc[9];
    v8f zero = {0.f, 0.f, 0.f, 0.f, 0.f, 0.f, 0.f, 0.f};
    #pragma unroll
    for (int j = 0; j < 9; ++j) acc[j] = zero;

    for (int kk = 0; kk < D_DIM; kk += 32) {
        __syncthreads();                  // prior tile fully consumed
        // ---- async DMA: x tile, 64x32 f32 = 512 float4, 4 per thread ----
        #pragma unroll
        for (int q = 0; q < 4; ++q) {
            int idx = q * 128 + tid;      // float4 index
            int row = idx >> 3, c4 = (idx & 7) * 4;
            const float* gp = x + (size_t)(row0 + row) * D_DIM + kk + c4;
            async_load_b128((unsigned)(size_t)&Xs[row][c4], gp);
        }
        // ---- async DMA: V tile, 32x144 f32 = 1152 float4, 9 per thread ----
        #pragma unroll
        for (int q = 0; q < 9; ++q) {
            int idx = (q * 128 + tid) * 4;
            const float* gp = V + (size_t)kk * KP + idx;
            async_load_b128((unsigned)(size_t)&Vs[idx], gp);
        }
        wait_async0();
        __syncthreads();

        // ---- s1 quadratic pass from LDS (2 threads per row) ----
        #pragma unroll
        for (int c4 = 0; c4 < 16; c4 += 4) {
            float4 xv = *(const float4*)&Xs[srow][cseg + c4];
            float vv[4] = {xv.x, xv.y, xv.z, xv.w};
            #pragma unroll
            for (int q = 0; q < 4; ++q) {
                int col = kk + cseg + c4 + q;
                float v = vv[q] - mu[col];
                s1acc += v * v * invsig[col];
            }
        }

        // ---- WMMA: wave w owns rows 16w..16w+15, all 9 col tiles ----
        for (int ks = 0; ks < 32; ks += 4) {
            int kloc = ks + 2 * half;
            v2f a;
            a.x = Xs[16 * wave + lan][kloc];
            a.y = Xs[16 * wave + lan][kloc + 1];
            #pragma unroll
            for (int j = 0; j < 9; ++j) {
                v2f b;
                b.x = Vs[kloc * KP + 16 * j + lan];
                b.y = Vs[(kloc + 1) * KP + 16 * j + lan];
                acc[j] = wmma_f32(a, b, acc[j]);
            }
        }
    }
    // ---- epilogue: T = acc - tmu (exact centering via linearity) ----
    #pragma unroll
    for (int j = 0; j < 9; ++j) {
        float tm = tmu[16 * j + lan];
        #pragma unroll
        for (int r = 0; r < 8; ++r) {
            int grow = row0 + 16 * wave + r + 8 * half;
            T[(size_t)grow * KP + 16 * j + lan] = acc[j][r] - tm;
        }
    }
    // ---- reduce s1 per row (2 threads/row) ----
    s1p[tid] = s1acc;
    __syncthreads();
    if (tid < 64) s1[row0 + tid] = s1p[2 * tid] + s1p[2 * tid + 1];
}

// ---------------------------------------------------------------------------
// K5: logp[n] = -0.5*(D*log2pi + logdetSigma + s1[n] - t^T Minv t)
// ---------------------------------------------------------------------------
__global__ __launch_bounds__(256) void k_corr(const float* __restrict__ T,
                                              const float* __restrict__ s1,
                                              const float* __restrict__ Minv,
                                              const float* __restrict__ scal,
                                              float* __restrict__ logp) {
    __shared__ float tsh[8][KP];
    int tid = threadIdx.x, lane = tid & 31, wave = tid >> 5;
    int n0 = blockIdx.x * 8;
    for (int i = tid; i < 8 * KP; i += 256) tsh[i / KP][i % KP] = T[(size_t)n0 * KP + i];
    __syncthreads();

    float z[5] = {0.f, 0.f, 0.f, 0.f, 0.f};
    for (int i = 0; i < KP; ++i) {
        float ti = tsh[wave][i];
        #pragma unroll
        for (int m = 0; m < 5; ++m) {
            int j = lane + 32 * m;
            if (j < KP) z[m] += ti * Minv[i * KP + j];
        }
    }
    float c = 0.f;
    #pragma unroll
    for (int m = 0; m < 5; ++m) {
        int j = lane + 32 * m;
        if (j < KP) c += z[m] * tsh[wave][j];
    }
    for (int off = 16; off > 0; off >>= 1) c += __shfl_xor(c, off, 32);
    if (lane == 0) {
        int n = n0 + wave;
        const float LOG2PI = 1.8378770664093453f;
        float q = s1[n] - c;
        logp[n] = -0.5f * ((float)D_DIM * LOG2PI + scal[0] + q);
    }
}

// ---------------------------------------------------------------------------
// K6: deterministic tree reduction + scale by full_data_size / N
// ---------------------------------------------------------------------------
__global__ __launch_bounds__(256) void k_reduce(const float* __restrict__ logp,
                                                const int* __restrict__ fds,
                                                float* __restrict__ out) {
    __shared__ float red[256];
    int tid = threadIdx.x;
    float s = 0.f;
    for (int i = tid; i < N_DIM; i += 256) s += logp[i];
    red[tid] = s; __syncthreads();
    for (int k = 128; k > 0; k >>= 1) { if (tid < k) red[tid] += red[tid + k]; __syncthreads(); }
    if (tid == 0) out[0] = red[0] * ((float)fds[0] / (float)N_DIM);
}

// ---------------------------------------------------------------------------
extern "C" void kernel_launch(void* const* d_in, const int* in_sizes, int n_in,
                              void* d_out, int out_size, void* d_ws, size_t ws_size,
                              hipStream_t stream) {
    const float* x     = (const float*)d_in[0];
    const float* theta = (const float*)d_in[1];
    const int*   fds   = (const int*)d_in[2];
    float*       out   = (float*)d_out;

    // workspace layout (floats)
    float* V      = (float*)d_ws;                    // D*KP
    float* invsig = V + (size_t)D_DIM * KP;          // D
    float* M      = invsig + D_DIM;                  // KP*KP
    float* Minv   = M + KP * KP;                     // KP*KP
    float* T      = Minv + KP * KP;                  // N*KP
    float* s1     = T + (size_t)N_DIM * KP;          // N
    float* logp   = s1 + N_DIM;                      // N
    float* scal   = logp + N_DIM;                    // 1
    float* tmu    = scal + 1;                        // KP

    k_build_V<<<(D_DIM * KP + 255) / 256, 256, 0, stream>>>(theta, V, invsig);
    k_tmu<<<1, 256, 0, stream>>>(theta, V, tmu);
    k_build_M<<<21, 128, 0, stream>>>(theta, V, M);
    k_chol_inv<<<1, 256, 0, stream>>>(theta, M, Minv, scal);
    k_gemm<<<N_DIM / 64, 128, 0, stream>>>(x, theta, V, invsig, tmu, T, s1);
    k_corr<<<N_DIM / 8, 256, 0, stream>>>(T, s1, Minv, scal, logp);
    k_reduce<<<1, 256, 0, stream>>>(logp, fds, out);
}